// NNConv_prot_80900003987923
// MI455X (gfx1250) — compile-verified
//
#include <hip/hip_runtime.h>

typedef __attribute__((ext_vector_type(2))) float v2f;
typedef __attribute__((ext_vector_type(8))) float v8f;

#define NN   10000
#define EE   30000
#define FN   32
#define FE   8
#define EMBD 64
#define HIDD 16
#define NG   64

// ---------------------------------------------------------------- utilities
__global__ void zero_f32(float* __restrict__ p, int n) {
  int i = blockIdx.x * blockDim.x + threadIdx.x;
  if (i < n) p[i] = 0.f;
}

// ------------------------------------------------- edge MLP hidden layers
// h0[e,k] = relu(edge_attr[e] @ nn0_w1 + nn0_b1)  (and same for nn1 -> h1)
__global__ void edge_hidden_kernel(const float* __restrict__ ea,
                                   const float* __restrict__ w1a, const float* __restrict__ b1a,
                                   const float* __restrict__ w1b, const float* __restrict__ b1b,
                                   float* __restrict__ h0, float* __restrict__ h1) {
  int e = blockIdx.x * blockDim.x + threadIdx.x;
  if (e >= EE) return;
  float a[FE];
#pragma unroll
  for (int j = 0; j < FE; ++j) a[j] = ea[e * FE + j];
#pragma unroll
  for (int k = 0; k < HIDD; ++k) {
    float s0 = b1a[k], s1 = b1b[k];
#pragma unroll
    for (int j = 0; j < FE; ++j) {
      s0 += a[j] * w1a[j * HIDD + k];
      s1 += a[j] * w1b[j * HIDD + k];
    }
    h0[e * HIDD + k] = fmaxf(s0, 0.f);
    h1[e * HIDD + k] = fmaxf(s1, 0.f);
  }
}

// ------------------------------------------------- repack nn*_w2 for GEMM
// w2: [HIDD, IN*EMBD]   ->   wp: [IN, HIDD*EMBD], wp[i, k*64+o] = w2[k, i*64+o]
__global__ void repack_w2_kernel(const float* __restrict__ w2, float* __restrict__ wp, int IN) {
  int idx = blockIdx.x * blockDim.x + threadIdx.x;
  int total = IN * HIDD * EMBD;   // cols = 1024
  if (idx >= total) return;
  int i   = idx >> 10;            // HIDD*EMBD == 1024
  int rem = idx & 1023;
  int k   = rem >> 6;
  int o   = rem & 63;
  wp[idx] = w2[(size_t)k * ((size_t)IN * EMBD) + (size_t)i * EMBD + o];
}

// ------------------------------------------------- fp32 WMMA GEMM (row-major)
// C[M,NB] = A[M,K] @ B[K,NB];  M%16==0, K%4==0, NB%(16*NT)==0.
// One wave computes a 16 x (16*NT) strip: A fragment loaded once per k-step,
// NT independent accumulator chains (hides WMMA latency, no D->A/B hazards).
// V_WMMA_F32_16X16X4_F32 layout (wave32):
//   A 16x4: lanes 0-15 -> M=lane, lanes 16-31 -> M=lane-16; v0 -> K=2*hi, v1 -> K=2*hi+1
//   B 4x16: col = lane&15; v0 -> K=2*hi, v1 -> K=2*hi+1
//   C 16x16: vgpr r -> row r + 8*hi, col = lane&15
template <int K, int NB, int NT>
__global__ void wmma_gemm_f32(const float* __restrict__ A, const float* __restrict__ B,
                              float* __restrict__ C, int M) {
  const int lane   = threadIdx.x & 31;
  const int wavei  = threadIdx.x >> 5;
  const int wpb    = blockDim.x >> 5;
  const int groupsN = NB / (16 * NT);
  const int groups  = (M >> 4) * groupsN;
  int g = blockIdx.x * wpb + wavei;
  if (g >= groups) return;                      // wave-uniform: EXEC all-1s for WMMA
  const int tm  = (g / groupsN) << 4;
  const int tn  = (g % groupsN) * (16 * NT);
  const int hi  = lane >> 4;
  const int l15 = lane & 15;

  const float* ap = A + (size_t)(tm + l15) * K + 2 * hi;
  const float* bp = B + (size_t)(2 * hi) * NB + tn + l15;

  v8f acc[NT];
#pragma unroll
  for (int t = 0; t < NT; ++t) acc[t] = (v8f){};

#pragma unroll
  for (int k0 = 0; k0 < K; k0 += 4) {
    v2f a;
    a.x = ap[k0];
    a.y = ap[k0 + 1];
#pragma unroll
    for (int t = 0; t < NT; ++t) {
      v2f b;
      b.x = bp[(size_t)k0 * NB + 16 * t];
      b.y = bp[(size_t)(k0 + 1) * NB + 16 * t];
      acc[t] = __builtin_amdgcn_wmma_f32_16x16x4_f32(false, a, false, b, (short)0,
                                                     acc[t], false, false);
    }
  }
#pragma unroll
  for (int t = 0; t < NT; ++t)
#pragma unroll
    for (int r = 0; r < 8; ++r)
      C[(size_t)(tm + r + 8 * hi) * NB + tn + 16 * t + l15] = acc[t][r];
}

// ------------------------------------------------- XB[n,o] = sum_i x[n,i]*b2[i*64+o]
__global__ void xb_kernel(const float* __restrict__ x, const float* __restrict__ b2r,
                          float* __restrict__ xb, int K) {
  int idx = blockIdx.x * blockDim.x + threadIdx.x;
  if (idx >= NN * EMBD) return;
  int n = idx >> 6, o = idx & 63;
  float s = 0.f;
  for (int i = 0; i < K; ++i) s += x[(size_t)n * K + i] * b2r[(size_t)i * EMBD + o];
  xb[idx] = s;
}

// ------------------------------------------------- per-edge message + scatter-add
// one wave32 per edge; lane owns channel pair (2*lane, 2*lane+1) -> b64 loads
__global__ void edge_combine_kernel(const int* __restrict__ src, const int* __restrict__ dst,
                                    const float* __restrict__ h, const float* __restrict__ Y,
                                    const float* __restrict__ XB, float* __restrict__ agg) {
  int e = (blockIdx.x * blockDim.x + threadIdx.x) >> 5;
  int lane = threadIdx.x & 31;
  if (e >= EE) return;
  int s = src[e], d = dst[e];
  const float* hrow = h + (size_t)e * HIDD;                     // wave-uniform address
  const v2f* y2  = (const v2f*)(Y + (size_t)s * (HIDD * EMBD)); // wave-uniform base
  const v2f* xb2 = (const v2f*)(XB + (size_t)s * EMBD);
  v2f acc = xb2[lane];
#pragma unroll
  for (int k = 0; k < HIDD; ++k) {
    float hk = hrow[k];
    v2f yv = y2[k * 32 + lane];                                 // global_load_b64, coalesced
    acc.x += hk * yv.x;
    acc.y += hk * yv.y;
  }
  float* arow = agg + (size_t)d * EMBD + 2 * lane;
  atomicAdd(&arow[0], acc.x);                                   // global_atomic_add_f32
  atomicAdd(&arow[1], acc.y);
}

// ------------------------------------------------- x = relu(agg + x@root + bias)
__global__ void node_update_kernel(const float* __restrict__ agg, const float* __restrict__ R,
                                   const float* __restrict__ bias, float* __restrict__ x) {
  int idx = blockIdx.x * blockDim.x + threadIdx.x;
  if (idx >= NN * EMBD) return;
  int o = idx & 63;
  x[idx] = fmaxf(agg[idx] + R[idx] + bias[o], 0.f);
}

// ------------------------------------------------- segment max-pool (x >= 0 post-ReLU)
__global__ void pool_max_kernel(const float* __restrict__ x, const int* __restrict__ batch,
                                float* __restrict__ pooled) {
  int idx = blockIdx.x * blockDim.x + threadIdx.x;
  if (idx >= NN * EMBD) return;
  int n = idx >> 6, o = idx & 63;
  // valid for non-negative floats: int ordering == float ordering; pooled pre-zeroed
  atomicMax((int*)&pooled[(size_t)batch[n] * EMBD + o], __float_as_int(x[idx]));
}

// ------------------------------------------------- head: [64,64]@[64,64]+b, then @[64,1]+b
__global__ void head_kernel(const float* __restrict__ pooled,
                            const float* __restrict__ w0, const float* __restrict__ b0,
                            const float* __restrict__ w1, const float* __restrict__ b1,
                            float* __restrict__ out) {
  int g = threadIdx.x;
  if (g >= NG) return;
  float o = 0.f;
  for (int j = 0; j < EMBD; ++j) {
    float hv = b0[j];
    for (int i = 0; i < EMBD; ++i) hv += pooled[(size_t)g * EMBD + i] * w0[(size_t)i * EMBD + j];
    o += hv * w1[j];
  }
  out[g] = o + b1[0];
}

// ==========================================================================
extern "C" void kernel_launch(void* const* d_in, const int* in_sizes, int n_in,
                              void* d_out, int out_size, void* d_ws, size_t ws_size,
                              hipStream_t stream) {
  const float* x_p    = (const float*)d_in[0];
  const float* eattr  = (const float*)d_in[2];
  const int*   eidx   = (const int*)d_in[4];
  const int*   batch  = (const int*)d_in[5];
  const float* nn0_w1 = (const float*)d_in[6];
  const float* nn0_b1 = (const float*)d_in[7];
  const float* nn0_w2 = (const float*)d_in[8];
  const float* nn0_b2 = (const float*)d_in[9];
  const float* nn1_w1 = (const float*)d_in[10];
  const float* nn1_b1 = (const float*)d_in[11];
  const float* nn1_w2 = (const float*)d_in[12];
  const float* nn1_b2 = (const float*)d_in[13];
  const float* root0  = (const float*)d_in[14];
  const float* bias0  = (const float*)d_in[15];
  const float* root1  = (const float*)d_in[16];
  const float* bias1  = (const float*)d_in[17];
  const float* root2  = (const float*)d_in[18];
  const float* bias2  = (const float*)d_in[19];
  const float* lin0_w = (const float*)d_in[20];
  const float* lin0_b = (const float*)d_in[21];
  const float* lin1_w = (const float*)d_in[22];
  const float* lin1_b = (const float*)d_in[23];
  float* out = (float*)d_out;

  // ---- workspace carve-up (fp32), ~56 MB total; Y (41 MB) stays L2-resident
  float* W = (float*)d_ws;
  size_t off = 0;
  auto carve = [&](size_t n) { float* p = W + off; off += n; return p; };
  float* h0   = carve((size_t)EE * HIDD);
  float* h1   = carve((size_t)EE * HIDD);
  float* W0p  = carve((size_t)FN * HIDD * EMBD);
  float* W1p  = carve((size_t)EMBD * HIDD * EMBD);
  float* Y    = carve((size_t)NN * HIDD * EMBD);
  float* XB   = carve((size_t)NN * EMBD);
  float* R    = carve((size_t)NN * EMBD);
  float* X    = carve((size_t)NN * EMBD);
  float* AGG  = carve((size_t)NN * EMBD);
  float* POOL = carve((size_t)NG * EMBD);

  const int* src = eidx;
  const int* dst = eidx + EE;

  const int NE_BLK = (NN * EMBD + 255) / 256;

  // edge-MLP hiddens for both MLPs, and weight repacks (once)
  edge_hidden_kernel<<<(EE + 255) / 256, 256, 0, stream>>>(eattr, nn0_w1, nn0_b1,
                                                           nn1_w1, nn1_b1, h0, h1);
  repack_w2_kernel<<<(FN * HIDD * EMBD + 255) / 256, 256, 0, stream>>>(nn0_w2, W0p, FN);
  repack_w2_kernel<<<(EMBD * HIDD * EMBD + 255) / 256, 256, 0, stream>>>(nn1_w2, W1p, EMBD);

  // wave-group counts: Y GEMMs use 16x64 strips (NT=4); root GEMMs 16x64 (NT=4, one group/row)
  const int grpY = (NN / 16) * ((HIDD * EMBD) / 64);   // 625 * 16 = 10000 waves
  const int grpR = (NN / 16);                          // 625 waves

  // ---- conv 0 (input x_p, K=32)
  wmma_gemm_f32<FN, HIDD * EMBD, 4><<<(grpY + 3) / 4, 128, 0, stream>>>(x_p, W0p, Y, NN);
  xb_kernel<<<NE_BLK, 256, 0, stream>>>(x_p, nn0_b2, XB, FN);
  wmma_gemm_f32<FN, EMBD, 4><<<(grpR + 3) / 4, 128, 0, stream>>>(x_p, root0, R, NN);
  zero_f32<<<NE_BLK, 256, 0, stream>>>(AGG, NN * EMBD);
  edge_combine_kernel<<<EE / 8, 256, 0, stream>>>(src, dst, h0, Y, XB, AGG);
  node_update_kernel<<<NE_BLK, 256, 0, stream>>>(AGG, R, bias0, X);

  // ---- convs 1,2 (shared edge MLP -> shared W1p; K=64)
  const float* roots[2]  = {root1, root2};
  const float* biases[2] = {bias1, bias2};
  for (int l = 0; l < 2; ++l) {
    wmma_gemm_f32<EMBD, HIDD * EMBD, 4><<<(grpY + 3) / 4, 128, 0, stream>>>(X, W1p, Y, NN);
    xb_kernel<<<NE_BLK, 256, 0, stream>>>(X, nn1_b2, XB, EMBD);
    wmma_gemm_f32<EMBD, EMBD, 4><<<(grpR + 3) / 4, 128, 0, stream>>>(X, roots[l], R, NN);
    zero_f32<<<NE_BLK, 256, 0, stream>>>(AGG, NN * EMBD);
    edge_combine_kernel<<<EE / 8, 256, 0, stream>>>(src, dst, h1, Y, XB, AGG);
    node_update_kernel<<<NE_BLK, 256, 0, stream>>>(AGG, R, biases[l], X);
  }

  // ---- pool + head
  zero_f32<<<(NG * EMBD + 255) / 256, 256, 0, stream>>>(POOL, NG * EMBD);
  pool_max_kernel<<<NE_BLK, 256, 0, stream>>>(X, batch, POOL);
  head_kernel<<<1, 64, 0, stream>>>(POOL, lin0_w, lin0_b, lin1_w, lin1_b, out);
}